// PixelBandwidth_35089882808471
// MI455X (gfx1250) — compile-verified
//
#include <hip/hip_runtime.h>
#include <cstdint>
#include <cstddef>

constexpr int BROWS = 2048;
constexpr int SLEN  = 512;
constexpr int KST   = SLEN - 1;              // 511 steps
constexpr int CHL   = 16;                    // chunk length
constexpr int NCH   = (KST + CHL - 1) / CHL; // 32 chunks
constexpr int NPAIR = NCH / 2;               // 16 composed pair-transforms
constexpr long long TOTAL_PAIRS = (long long)BROWS * NPAIR;  // 32768

#if defined(__has_builtin)
#if __has_builtin(__builtin_amdgcn_global_load_async_to_lds_b128) && \
    __has_builtin(__builtin_amdgcn_s_wait_asynccnt)
#define USE_ASYNC_LDS 1
#endif
#if __has_builtin(__builtin_amdgcn_wmma_f32_16x16x4_f32)
#define USE_WMMA 1
#endif
#endif
#ifndef USE_ASYNC_LDS
#define USE_ASYNC_LDS 0
#endif
#ifndef USE_WMMA
#define USE_WMMA 0
#endif

#if USE_ASYNC_LDS
typedef int v4i __attribute__((__vector_size__(4 * sizeof(int))));
typedef __attribute__((address_space(1))) v4i* glob_v4i_ptr;
typedef __attribute__((address_space(3))) v4i* lds_v4i_ptr;
#endif
#if USE_WMMA
typedef float v2f __attribute__((ext_vector_type(2)));
typedef float v8f __attribute__((ext_vector_type(8)));
#endif

__device__ __forceinline__ float softplusf(float x) { return log1pf(expf(x)); }

struct Params {
  float tau_in_prod, tau_mil_prod, Aamp1, Aloop1, tau_out, osf, odf;
};

__device__ __forceinline__ Params load_params(const float* s0, const float* s1,
                                              const float* s2, const float* s3,
                                              const float* s4, const float* s5,
                                              const float* s6) {
  Params P;
  P.tau_in_prod  = s0[0];
  P.tau_mil_prod = softplusf(s1[0]);
  P.Aamp1        = 1.0f / softplusf(s2[0]) + 1.0f;   // A_amp + 1
  P.Aloop1       = 1.0f / softplusf(s3[0]) + 1.0f;   // A_loop + 1
  P.tau_out      = softplusf(s4[0]);
  P.osf          = 1.0f / softplusf(s5[0]);
  P.odf          = 1.0f / softplusf(s6[0]);
  return P;
}

// Z(4x6) = X(4x6, rows 4,5 implicitly zero) * Y(4x6, rows 4,5 implicitly zero)
__device__ __forceinline__ void mul46(const float (&X)[4][6], const float (&Y)[4][6],
                                      float (&Z)[4][6]) {
#pragma unroll
  for (int i = 0; i < 4; ++i) {
#pragma unroll
    for (int j = 0; j < 6; ++j) {
      Z[i][j] = fmaf(X[i][0], Y[0][j],
                fmaf(X[i][1], Y[1][j],
                fmaf(X[i][2], Y[2][j], X[i][3] * Y[3][j])));
    }
  }
}

// exp(em) rows 0..3 where em is the reference's 6x6 FOH matrix.
// em row4 = e5 (value 1.0), em row5 = 0 => every power >=2 has rows 4,5 == 0;
// the Pade rational keeps row5 = e5, row4 = e4 + tau*e5. We carry only the top
// 4x6 block + scalar tau. Pade-13 + scaling-squaring.
__device__ __forceinline__ void expm_foh(float a, float w, float osf, float odf,
                                         float dt, float (&R)[4][6]) {
  float M[4][6];
#pragma unroll
  for (int i = 0; i < 4; ++i)
#pragma unroll
    for (int j = 0; j < 6; ++j) M[i][j] = 0.0f;
  M[0][0] = -a * dt;  M[0][1] = -w * dt;  M[0][4] = w * dt;
  M[1][0] = dt;
  M[2][1] = osf * dt; M[2][2] = -osf * dt;
  M[3][2] = odf * dt; M[3][3] = -odf * dt;

  float nrm = 0.0f;
#pragma unroll
  for (int j = 0; j < 6; ++j) {
    float cs = fabsf(M[0][j]) + fabsf(M[1][j]) + fabsf(M[2][j]) + fabsf(M[3][j]);
    if (j == 5) cs += 1.0f;
    nrm = fmaxf(nrm, cs);
  }

  const float theta13 = 5.3719203511481517f;
  int s = 0;
  if (nrm > theta13) {
    s = (int)ceilf(log2f(nrm / theta13));
    s = s < 0 ? 0 : (s > 30 ? 30 : s);
  }
  const float sigma = exp2f((float)(-s));
#pragma unroll
  for (int i = 0; i < 4; ++i)
#pragma unroll
    for (int j = 0; j < 6; ++j) M[i][j] *= sigma;

  float M2[4][6], M4[4][6], M6[4][6];
  mul46(M, M, M2);
#pragma unroll
  for (int i = 0; i < 4; ++i) M2[i][5] = fmaf(M[i][4], sigma, M2[i][5]);
  mul46(M2, M2, M4);
  mul46(M4, M2, M6);

  const float b0  = 6.476475253248e16f,  b1  = 3.238237626624e16f;
  const float b2  = 7.7717703038976e15f, b3  = 1.1873537964288e15f;
  const float b4  = 1.29060195264e14f,   b5  = 1.05594705216e13f;
  const float b6  = 6.704425728e11f,     b7  = 3.352212864e10f;
  const float b8  = 1.32324192e9f,       b9  = 4.08408e7f;
  const float b10 = 9.6096e5f,           b11 = 1.6380e4f;
  const float b12 = 1.82e2f,             b13 = 1.0f;

  float T1[4][6], W[4][6], U[4][6], V[4][6];
#pragma unroll
  for (int i = 0; i < 4; ++i)
#pragma unroll
    for (int j = 0; j < 6; ++j)
      T1[i][j] = fmaf(b13, M6[i][j], fmaf(b11, M4[i][j], b9 * M2[i][j]));
  mul46(M6, T1, W);
#pragma unroll
  for (int i = 0; i < 4; ++i)
#pragma unroll
    for (int j = 0; j < 6; ++j)
      W[i][j] = W[i][j] + fmaf(b7, M6[i][j], fmaf(b5, M4[i][j], b3 * M2[i][j]));
#pragma unroll
  for (int i = 0; i < 4; ++i)
#pragma unroll
    for (int j = 0; j < 6; ++j)
      U[i][j] = fmaf(M[i][0], W[0][j], fmaf(M[i][1], W[1][j],
                fmaf(M[i][2], W[2][j], fmaf(M[i][3], W[3][j], b1 * M[i][j]))));

#pragma unroll
  for (int i = 0; i < 4; ++i)
#pragma unroll
    for (int j = 0; j < 6; ++j)
      T1[i][j] = fmaf(b12, M6[i][j], fmaf(b10, M4[i][j], b8 * M2[i][j]));
  mul46(M6, T1, W);
#pragma unroll
  for (int i = 0; i < 4; ++i)
#pragma unroll
    for (int j = 0; j < 6; ++j) {
      float v = W[i][j] + fmaf(b6, M6[i][j], fmaf(b4, M4[i][j], b2 * M2[i][j]));
      if (j == i) v += b0;
      V[i][j] = v;
    }

  float G[4][10];
#pragma unroll
  for (int i = 0; i < 4; ++i) {
#pragma unroll
    for (int j = 0; j < 4; ++j) G[i][j] = V[i][j] - U[i][j];
#pragma unroll
    for (int j = 0; j < 6; ++j) G[i][4 + j] = V[i][j] + U[i][j];
    float q4 = V[i][4] - U[i][4];
    float q5 = V[i][5] - U[i][5];
    G[i][8] -= q4;
    G[i][9] -= fmaf(sigma, q4, q5);
  }

#pragma unroll
  for (int col = 0; col < 4; ++col) {
#pragma unroll
    for (int r = col + 1; r < 4; ++r) {
      bool sw = fabsf(G[r][col]) > fabsf(G[col][col]);
#pragma unroll
      for (int j = 0; j < 10; ++j) {
        float hi = G[col][j], lo = G[r][j];
        G[col][j] = sw ? lo : hi;
        G[r][j]   = sw ? hi : lo;
      }
    }
    float inv = 1.0f / G[col][col];
#pragma unroll
    for (int r = 0; r < 4; ++r) {
      if (r == col) continue;
      float f = G[r][col] * inv;
#pragma unroll
      for (int j = 0; j < 10; ++j) G[r][j] = fmaf(-f, G[col][j], G[r][j]);
    }
  }
#pragma unroll
  for (int i = 0; i < 4; ++i) {
    float inv = 1.0f / G[i][i];
#pragma unroll
    for (int j = 0; j < 6; ++j) R[i][j] = G[i][4 + j] * inv;
  }

  float tau = sigma;
  for (int it = 0; it < s; ++it) {
    float Rn[4][6];
    mul46(R, R, Rn);
#pragma unroll
    for (int i = 0; i < 4; ++i) {
      Rn[i][4] += R[i][4];
      Rn[i][5] = fmaf(tau, R[i][4], Rn[i][5] + R[i][5]);
    }
#pragma unroll
    for (int i = 0; i < 4; ++i)
#pragma unroll
      for (int j = 0; j < 6; ++j) R[i][j] = Rn[i][j];
    tau += tau;
  }
}

__device__ __forceinline__ void step_mats(const Params& P, float Iss, float dt,
                                          float (&R)[4][6]) {
  float tau_in  = P.tau_in_prod / Iss;
  float tau_mil = P.tau_mil_prod / Iss;
  float inv_denom = 1.0f / ((tau_in + tau_mil) * P.tau_out);
  float a = (tau_in + P.tau_out + P.Aamp1 * tau_mil) * inv_denom;
  float w = P.Aloop1 * inv_denom;
  expm_foh(a, w, P.osf, P.odf, dt, R);
}

// Phase 1: per (row, chunk) compose the affine transforms of CHL steps.
__global__ __launch_bounds__(128) void k_compose(
    const float* __restrict__ it_eff, const float* __restrict__ ts,
    const float* s0, const float* s1, const float* s2, const float* s3,
    const float* s4, const float* s5, const float* s6,
    float* __restrict__ chunkT) {
  int t = blockIdx.x * blockDim.x + threadIdx.x;
  if (t >= BROWS * NCH) return;
  int b = t / NCH, c = t % NCH;
  Params P = load_params(s0, s1, s2, s3, s4, s5, s6);
  int k0 = c * CHL;
  int k1 = k0 + CHL; if (k1 > KST) k1 = KST;
  const float* itr = it_eff + (size_t)b * SLEN;
  const float* tsr = ts     + (size_t)b * SLEN;

  float A[4][4] = {{1,0,0,0},{0,1,0,0},{0,0,1,0},{0,0,0,1}};
  float cv[4] = {0.f, 0.f, 0.f, 0.f};
  float it_p = itr[k0], ts_p = tsr[k0], u_p = logf(it_p);
  for (int k = k0; k < k1; ++k) {
    __builtin_prefetch(itr + k + 2, 0, 0);
    float it_n = itr[k + 1], ts_n = tsr[k + 1];
    float dt = ts_n - ts_p;
    float u_n = logf(it_n);
    float R[4][6];
    step_mats(P, it_n, dt, R);
    float ck[4];
#pragma unroll
    for (int i = 0; i < 4; ++i)
      ck[i] = fmaf(R[i][4] - R[i][5], u_p, R[i][5] * u_n);
    float An[4][4], cn[4];
#pragma unroll
    for (int i = 0; i < 4; ++i) {
#pragma unroll
      for (int j = 0; j < 4; ++j)
        An[i][j] = fmaf(R[i][0], A[0][j], fmaf(R[i][1], A[1][j],
                   fmaf(R[i][2], A[2][j], R[i][3] * A[3][j])));
      cn[i] = fmaf(R[i][0], cv[0], fmaf(R[i][1], cv[1],
              fmaf(R[i][2], cv[2], fmaf(R[i][3], cv[3], ck[i]))));
    }
#pragma unroll
    for (int i = 0; i < 4; ++i) {
      cv[i] = cn[i];
#pragma unroll
      for (int j = 0; j < 4; ++j) A[i][j] = An[i][j];
    }
    ts_p = ts_n; u_p = u_n;
  }
  float* out = chunkT + (size_t)t * 20;
#pragma unroll
  for (int i = 0; i < 4; ++i)
#pragma unroll
    for (int j = 0; j < 4; ++j) out[i * 4 + j] = A[i][j];
#pragma unroll
  for (int i = 0; i < 4; ++i) out[16 + i] = cv[i];
}

// ---- Phase 1.5: pairwise composition of chunk transforms -------------------
// P_pair = T_{2q+1} o T_{2q} as 5x5 homogeneous products. WMMA variant packs
// three 5x5 products block-diagonally into one 16x16 tile and runs a K=16
// matmul as 4 accumulate-chained V_WMMA_F32_16X16X4_F32 ops.

__device__ __forceinline__ float hom_elem(const float* __restrict__ chunkT,
                                          long long slot, int r, int c) {
  if (r < 4) {
    if (c < 4) return chunkT[slot * 20 + r * 4 + c];
    return chunkT[slot * 20 + 16 + r];
  }
  return (c == 4) ? 1.0f : 0.0f;
}

#if USE_WMMA
__device__ __forceinline__ float big_elem(const float* __restrict__ chunkT,
                                          const long long (&slot)[3],
                                          const bool (&valid)[3],
                                          int row, int col) {
  if (row >= 15 || col >= 15) return 0.0f;
  int j = row / 5;
  if (col / 5 != j) return 0.0f;
  if (!valid[j]) return 0.0f;
  return hom_elem(chunkT, slot[j], row - 5 * j, col - 5 * j);
}

__global__ __launch_bounds__(128) void k_paircompose(
    const float* __restrict__ chunkT, float* __restrict__ chunkP) {
  int gtid = blockIdx.x * blockDim.x + threadIdx.x;
  int wave = gtid >> 5;
  int lane = threadIdx.x & 31;
  int half = lane >> 4;
  int lm = lane & 15;

  long long g0 = (long long)wave * 3;
  bool valid[3];
  long long sLo[3], sHi[3];
#pragma unroll
  for (int j = 0; j < 3; ++j) {
    long long p = g0 + j;
    bool v = p < TOTAL_PAIRS;
    valid[j] = v;
    long long pb = v ? (p / NPAIR) : 0;
    long long pq = v ? (p % NPAIR) : 0;
    sLo[j] = pb * NCH + 2 * pq;   // chunk applied first
    sHi[j] = sLo[j] + 1;          // chunk applied second
  }

  v8f acc = {0.f, 0.f, 0.f, 0.f, 0.f, 0.f, 0.f, 0.f};
#pragma unroll
  for (int k = 0; k < 4; ++k) {
    // A 16x4 slice: lane m holds (m, 4k+{0,1}) low half, (m, 4k+{2,3}) high half
    // B 4x16 slice: lane n holds (4k+{0,1}, n) low half, (4k+{2,3}, n) high half
    int ka = 4 * k + (half ? 2 : 0);
    int kb = 4 * k + (half ? 3 : 1);
    v2f af, bf;
    af.x = big_elem(chunkT, sHi, valid, lm, ka);
    af.y = big_elem(chunkT, sHi, valid, lm, kb);
    bf.x = big_elem(chunkT, sLo, valid, ka, lm);
    bf.y = big_elem(chunkT, sLo, valid, kb, lm);
    acc = __builtin_amdgcn_wmma_f32_16x16x4_f32(false, af, false, bf,
                                                (short)0, acc, false, false);
  }

  // D: VGPR r holds row r (lanes 0-15) / row r+8 (lanes 16-31), col = lane%16.
#pragma unroll
  for (int r = 0; r < 8; ++r) {
    int rr = r + 8 * half;
    int cc = lm;
    if (rr < 15 && cc < 15) {
      int j = rr / 5;
      if (cc / 5 == j && valid[j]) {
        int br = rr - 5 * j, bc = cc - 5 * j;
        long long p = g0 + j;
        if (br < 4) {
          if (bc < 4)       chunkP[p * 20 + br * 4 + bc] = acc[r];
          else if (bc == 4) chunkP[p * 20 + 16 + br]     = acc[r];
        }
      }
    }
  }
}
#else
__global__ __launch_bounds__(128) void k_paircompose(
    const float* __restrict__ chunkT, float* __restrict__ chunkP) {
  long long p = (long long)blockIdx.x * blockDim.x + threadIdx.x;
  if (p >= TOTAL_PAIRS) return;
  long long pb = p / NPAIR, pq = p % NPAIR;
  const float* L = chunkT + (pb * NCH + 2 * pq) * 20;
  const float* H = L + 20;
  float* O = chunkP + p * 20;
#pragma unroll
  for (int i = 0; i < 4; ++i) {
#pragma unroll
    for (int j = 0; j < 4; ++j)
      O[i * 4 + j] = fmaf(H[i * 4 + 0], L[0 * 4 + j], fmaf(H[i * 4 + 1], L[1 * 4 + j],
                     fmaf(H[i * 4 + 2], L[2 * 4 + j], H[i * 4 + 3] * L[3 * 4 + j])));
    O[16 + i] = fmaf(H[i * 4 + 0], L[16 + 0], fmaf(H[i * 4 + 1], L[16 + 1],
                fmaf(H[i * 4 + 2], L[16 + 2], fmaf(H[i * 4 + 3], L[16 + 3], H[16 + i]))));
  }
}
#endif

// Phase 2: per-row serial scan over NPAIR composed transforms (critical path).
// Transform stream double-buffered into LDS with gfx1250 async loads.
__global__ __launch_bounds__(256) void k_scan(
    const float* __restrict__ it_eff,
    const float* __restrict__ chunkP,
    float* __restrict__ chunkX,
    float* __restrict__ y) {
  int b = blockIdx.x * blockDim.x + threadIdx.x;
  if (b >= BROWS) return;
  int tid = threadIdx.x;
  (void)tid;
  float u0 = logf(it_eff[(size_t)b * SLEN]);
  float x[4] = {0.f, u0, u0, u0};
  y[(size_t)b * SLEN] = u0;
  const float* gbase = chunkP + (size_t)b * NPAIR * 20;

#if USE_ASYNC_LDS
  __shared__ __align__(16) float sbuf[2][256 * 20];
  {
    float* l = &sbuf[0][tid * 20];
#pragma unroll
    for (int q = 0; q < 5; ++q)
      __builtin_amdgcn_global_load_async_to_lds_b128(
          (glob_v4i_ptr)(gbase + q * 4),
          (lds_v4i_ptr)(l + q * 4), 0, 0);
  }
#endif

  for (int c = 0; c < NPAIR; ++c) {
#pragma unroll
    for (int i = 0; i < 4; ++i)
      chunkX[((size_t)b * NPAIR + c) * 4 + i] = x[i];

    float T[20];
#if USE_ASYNC_LDS
    if (c + 1 < NPAIR) {
      const float* gn = gbase + (size_t)(c + 1) * 20;
      float* l = &sbuf[(c + 1) & 1][tid * 20];
#pragma unroll
      for (int q = 0; q < 5; ++q)
        __builtin_amdgcn_global_load_async_to_lds_b128(
            (glob_v4i_ptr)(gn + q * 4),
            (lds_v4i_ptr)(l + q * 4), 0, 0);
      __builtin_amdgcn_s_wait_asynccnt(5);  // loads complete in order
    } else {
      __builtin_amdgcn_s_wait_asynccnt(0);
    }
    {
      const float* l = &sbuf[c & 1][tid * 20];
#pragma unroll
      for (int q = 0; q < 20; ++q) T[q] = l[q];
    }
#else
    if (c + 1 < NPAIR) __builtin_prefetch(gbase + (size_t)(c + 1) * 20, 0, 0);
    const float* g = gbase + (size_t)c * 20;
#pragma unroll
    for (int q = 0; q < 20; ++q) T[q] = g[q];
#endif
    float xn[4];
#pragma unroll
    for (int i = 0; i < 4; ++i)
      xn[i] = fmaf(T[i * 4 + 0], x[0], fmaf(T[i * 4 + 1], x[1],
              fmaf(T[i * 4 + 2], x[2], fmaf(T[i * 4 + 3], x[3], T[16 + i]))));
#pragma unroll
    for (int i = 0; i < 4; ++i) x[i] = xn[i];
  }
}

// Phase 3: per (row, chunk) replay from the pair-boundary state, emit y[b][k+1].
__global__ __launch_bounds__(128) void k_emit(
    const float* __restrict__ it_eff, const float* __restrict__ ts,
    const float* s0, const float* s1, const float* s2, const float* s3,
    const float* s4, const float* s5, const float* s6,
    const float* __restrict__ chunkT,
    const float* __restrict__ chunkX,
    float* __restrict__ y) {
  int t = blockIdx.x * blockDim.x + threadIdx.x;
  if (t >= BROWS * NCH) return;
  int b = t / NCH, c = t % NCH;
  Params P = load_params(s0, s1, s2, s3, s4, s5, s6);
  int k0 = c * CHL;
  int k1 = k0 + CHL; if (k1 > KST) k1 = KST;
  const float* itr = it_eff + (size_t)b * SLEN;
  const float* tsr = ts     + (size_t)b * SLEN;

  // state at pair boundary 2*(c/2)
  float x[4];
#pragma unroll
  for (int i = 0; i < 4; ++i)
    x[i] = chunkX[((size_t)b * NPAIR + (c >> 1)) * 4 + i];
  if (c & 1) {
    // advance over the even chunk to reach the start of this odd chunk
    const float* Tp = chunkT + ((size_t)b * NCH + (c - 1)) * 20;
    float xn[4];
#pragma unroll
    for (int i = 0; i < 4; ++i)
      xn[i] = fmaf(Tp[i * 4 + 0], x[0], fmaf(Tp[i * 4 + 1], x[1],
              fmaf(Tp[i * 4 + 2], x[2], fmaf(Tp[i * 4 + 3], x[3], Tp[16 + i]))));
#pragma unroll
    for (int i = 0; i < 4; ++i) x[i] = xn[i];
  }

  float it_p = itr[k0], ts_p = tsr[k0], u_p = logf(it_p);
  for (int k = k0; k < k1; ++k) {
    __builtin_prefetch(itr + k + 2, 0, 0);
    float it_n = itr[k + 1], ts_n = tsr[k + 1];
    float dt = ts_n - ts_p;
    float u_n = logf(it_n);
    float R[4][6];
    step_mats(P, it_n, dt, R);
    float xn[4];
#pragma unroll
    for (int i = 0; i < 4; ++i)
      xn[i] = fmaf(R[i][0], x[0], fmaf(R[i][1], x[1],
              fmaf(R[i][2], x[2], fmaf(R[i][3], x[3],
              fmaf(R[i][4] - R[i][5], u_p, R[i][5] * u_n)))));
#pragma unroll
    for (int i = 0; i < 4; ++i) x[i] = xn[i];
    y[(size_t)b * SLEN + k + 1] = x[3];
    ts_p = ts_n; u_p = u_n;
  }
}

extern "C" void kernel_launch(void* const* d_in, const int* in_sizes, int n_in,
                              void* d_out, int out_size, void* d_ws, size_t ws_size,
                              hipStream_t stream) {
  (void)in_sizes; (void)n_in; (void)out_size; (void)ws_size;
  const float* it_eff = (const float*)d_in[0];
  const float* ts     = (const float*)d_in[1];
  const float* s0 = (const float*)d_in[2];  // tau_in_it_eff_prod
  const float* s1 = (const float*)d_in[3];  // tau_mil_raw
  const float* s2 = (const float*)d_in[4];  // A_amp_inv_raw
  const float* s3 = (const float*)d_in[5];  // A_loop_inv_raw
  const float* s4 = (const float*)d_in[6];  // tau_out_raw
  const float* s5 = (const float*)d_in[7];  // tau_sf_raw
  const float* s6 = (const float*)d_in[8];  // tau_diff_raw
  float* y = (float*)d_out;

  float* chunkT = (float*)d_ws;                         // [B][NCH][20]
  float* chunkP = chunkT + (size_t)BROWS * NCH * 20;    // [B][NPAIR][20]
  float* chunkX = chunkP + (size_t)BROWS * NPAIR * 20;  // [B][NPAIR][4]

  dim3 blk1(128), grd1((BROWS * NCH + 127) / 128);
  // 3 pair-compositions per wave in the WMMA variant
  long long waves = (TOTAL_PAIRS + 2) / 3;
  int pair_blocks = (int)((waves * 32 + 127) / 128);

  k_compose<<<grd1, blk1, 0, stream>>>(it_eff, ts, s0, s1, s2, s3, s4, s5, s6, chunkT);
  k_paircompose<<<dim3(pair_blocks), blk1, 0, stream>>>(chunkT, chunkP);
  k_scan<<<dim3(BROWS / 256), dim3(256), 0, stream>>>(it_eff, chunkP, chunkX, y);
  k_emit<<<grd1, blk1, 0, stream>>>(it_eff, ts, s0, s1, s2, s3, s4, s5, s6,
                                    chunkT, chunkX, y);
}